// SelectiveScan1D_27075473834177
// MI455X (gfx1250) — compile-verified
//
#include <hip/hip_runtime.h>
#include <hip/hip_bf16.h>

// ---------------------------------------------------------------------------
// Mamba selective scan, MI455X (gfx1250, wave32).
//   K1: x_dbl = x @ W_xproj^T              (fp32 WMMA 16x16x4)
//   K2: delta = softplus(dt_r @ W_dt^T+b)  (fp32 WMMA 16x16x4)
//   K3: per-(b,d,n,chunk) partial scan     (chunked over L for occupancy)
//   K4: carry combine across chunks
//   K5: re-scan with carries, y = sum_n h*C + x*D  (8-lane shfl reduction)
// ---------------------------------------------------------------------------

#define BSZ     2
#define LSEQ    2048
#define DIN     2048
#define DSTATE  8
#define DTRANK  256
#define KPROJ   (DTRANK + 2 * DSTATE)   // 272
#define NCHUNK  8
#define LCHUNK  (LSEQ / NCHUNK)         // 256

typedef float v2f __attribute__((ext_vector_type(2)));
typedef float v8f __attribute__((ext_vector_type(8)));

// ---------------- K1: x_dbl[row][col] = sum_k x[row][k] * W_xproj[col][k] ---
// One wave per 16x16 output tile. 256 row tiles x 17 col tiles = 4352 tiles.
__global__ __launch_bounds__(256) void proj_kernel(
    const float* __restrict__ x,        // (BSZ*LSEQ, DIN)
    const float* __restrict__ Wxp,      // (KPROJ, DIN)
    float* __restrict__ xdbl)           // (BSZ*LSEQ, KPROJ)
{
    const int lane = threadIdx.x & 31;
    const int wave = threadIdx.x >> 5;
    const int tile = blockIdx.x * 8 + wave;            // 544 blocks * 8 waves
    const int rt = tile / 17;                          // row tile
    const int ct = tile % 17;                          // col tile
    const int m  = lane & 15;
    const int hi = lane >> 4;

    const float* arow = x   + (size_t)(rt * 16 + m) * DIN + 2 * hi;
    const float* brow = Wxp + (size_t)(ct * 16 + m) * DIN + 2 * hi;

    v8f acc = {};
    for (int k = 0; k < DIN; k += 4) {
        v2f a = *(const v2f*)(arow + k);   // A[m][2hi], A[m][2hi+1]
        v2f b = *(const v2f*)(brow + k);   // B[2hi][n], B[2hi+1][n]  (n == m)
        acc = __builtin_amdgcn_wmma_f32_16x16x4_f32(
            false, a, false, b, (short)0, acc, false, false);
    }
    // D layout: VGPR j -> M = j + 8*hi, N = lane&15
    #pragma unroll
    for (int j = 0; j < 8; ++j) {
        const int row = rt * 16 + j + 8 * hi;
        const int col = ct * 16 + m;
        xdbl[(size_t)row * KPROJ + col] = acc[j];
    }
}

// ---------------- K2: delta = softplus(dt_r @ W_dt^T + b_dt) ---------------
// dt_r = xdbl[:, 0:256] (row stride KPROJ). Output (4096, 2048).
__global__ __launch_bounds__(256) void delta_kernel(
    const float* __restrict__ xdbl,     // (BSZ*LSEQ, KPROJ)
    const float* __restrict__ Wdt,      // (DIN, DTRANK)
    const float* __restrict__ b_dt,     // (DIN)
    float* __restrict__ delta)          // (BSZ*LSEQ, DIN)
{
    const int lane = threadIdx.x & 31;
    const int wave = threadIdx.x >> 5;
    const int tile = blockIdx.x * 8 + wave;            // 4096 blocks * 8 waves
    const int rt = tile >> 7;                          // 256 row tiles
    const int ct = tile & 127;                         // 128 col tiles
    const int m  = lane & 15;
    const int hi = lane >> 4;

    const float* arow = xdbl + (size_t)(rt * 16 + m) * KPROJ + 2 * hi;
    const float* brow = Wdt  + (size_t)(ct * 16 + m) * DTRANK + 2 * hi;

    v8f acc = {};
    for (int k = 0; k < DTRANK; k += 4) {
        v2f a = *(const v2f*)(arow + k);
        v2f b = *(const v2f*)(brow + k);
        acc = __builtin_amdgcn_wmma_f32_16x16x4_f32(
            false, a, false, b, (short)0, acc, false, false);
    }
    #pragma unroll
    for (int j = 0; j < 8; ++j) {
        const int row = rt * 16 + j + 8 * hi;
        const int col = ct * 16 + m;
        float v = acc[j] + b_dt[col];
        float sp = (v > 20.0f) ? v : log1pf(__expf(v));   // softplus
        delta[(size_t)row * DIN + col] = sp;
    }
}

// ---------------- K3: per-chunk partial scans ------------------------------
// thread t -> n = t&7, c = (t>>3)&7, d = (t>>6)&2047, b = t>>17
__global__ __launch_bounds__(256) void scan_pass1(
    const float* __restrict__ delta,
    const float* __restrict__ x,
    const float* __restrict__ xdbl,
    const float* __restrict__ A_log,
    float* __restrict__ aProd,          // (BSZ*DIN, NCHUNK, DSTATE)
    float* __restrict__ hPart)
{
    const int t = blockIdx.x * blockDim.x + threadIdx.x;
    const int n = t & 7;
    const int c = (t >> 3) & 7;
    const int d = (t >> 6) & (DIN - 1);
    const int b = t >> 17;

    const float A = -__expf(A_log[d * DSTATE + n]);
    float h = 0.0f, aP = 1.0f;
    const int l0 = c * LCHUNK;
    for (int i = 0; i < LCHUNK; ++i) {
        const int row = b * LSEQ + l0 + i;
        const float dv = delta[(size_t)row * DIN + d];
        const float xv = x[(size_t)row * DIN + d];
        const float Bv = xdbl[(size_t)row * KPROJ + DTRANK + n];
        const float dA = __expf(dv * A);
        h  = fmaf(dA, h, dv * xv * Bv);
        aP *= dA;
    }
    const int idx = (((b * DIN + d) * NCHUNK) + c) * DSTATE + n;
    aProd[idx] = aP;
    hPart[idx] = h;
}

// ---------------- K4: sequential carry combine over 8 chunks ---------------
__global__ __launch_bounds__(256) void scan_pass2(
    const float* __restrict__ aProd,
    const float* __restrict__ hPart,
    float* __restrict__ hIn)
{
    const int t = blockIdx.x * blockDim.x + threadIdx.x; // (b*DIN+d)*8+n
    const int n  = t & 7;
    const int bd = t >> 3;
    float h = 0.0f;
    #pragma unroll
    for (int c = 0; c < NCHUNK; ++c) {
        const int idx = (bd * NCHUNK + c) * DSTATE + n;
        hIn[idx] = h;
        h = fmaf(aProd[idx], h, hPart[idx]);
    }
}

// ---------------- K5: re-scan with carries, emit y -------------------------
__global__ __launch_bounds__(256) void scan_pass3(
    const float* __restrict__ delta,
    const float* __restrict__ x,
    const float* __restrict__ xdbl,
    const float* __restrict__ A_log,
    const float* __restrict__ D_param,
    const float* __restrict__ hIn,
    float* __restrict__ y)
{
    const int t = blockIdx.x * blockDim.x + threadIdx.x;
    const int n = t & 7;
    const int c = (t >> 3) & 7;
    const int d = (t >> 6) & (DIN - 1);
    const int b = t >> 17;

    const float A  = -__expf(A_log[d * DSTATE + n]);
    const float Dp = D_param[d];
    float h = hIn[(((b * DIN + d) * NCHUNK) + c) * DSTATE + n];
    const int l0 = c * LCHUNK;
    for (int i = 0; i < LCHUNK; ++i) {
        const int row = b * LSEQ + l0 + i;
        const float dv = delta[(size_t)row * DIN + d];
        const float xv = x[(size_t)row * DIN + d];
        const float Bv = xdbl[(size_t)row * KPROJ + DTRANK + n];
        const float Cv = xdbl[(size_t)row * KPROJ + DTRANK + DSTATE + n];
        const float dA = __expf(dv * A);
        h = fmaf(dA, h, dv * xv * Bv);
        float part = h * Cv;                 // reduce over n (8 lanes, wave32)
        part += __shfl_xor(part, 1, 32);
        part += __shfl_xor(part, 2, 32);
        part += __shfl_xor(part, 4, 32);
        if (n == 0)
            y[(size_t)row * DIN + d] = part + xv * Dp;
    }
}

// ---------------------------------------------------------------------------
extern "C" void kernel_launch(void* const* d_in, const int* in_sizes, int n_in,
                              void* d_out, int out_size, void* d_ws, size_t ws_size,
                              hipStream_t stream) {
    const float* x       = (const float*)d_in[0];
    const float* W_xproj = (const float*)d_in[1];
    const float* W_dt    = (const float*)d_in[2];
    const float* b_dt    = (const float*)d_in[3];
    const float* A_log   = (const float*)d_in[4];
    const float* D_param = (const float*)d_in[5];
    float* y = (float*)d_out;

    // Workspace carve-up (floats)
    float* ws    = (float*)d_ws;
    float* xdbl  = ws;                                   // 4096*272
    float* delta = xdbl + (size_t)BSZ * LSEQ * KPROJ;    // 4096*2048
    float* aProd = delta + (size_t)BSZ * LSEQ * DIN;     // 2*2048*8*8
    float* hPart = aProd + (size_t)BSZ * DIN * NCHUNK * DSTATE;
    float* hIn   = hPart + (size_t)BSZ * DIN * NCHUNK * DSTATE;

    // K1: 4352 tiles, 8 waves/block -> 544 blocks
    proj_kernel<<<544, 256, 0, stream>>>(x, W_xproj, xdbl);
    // K2: 32768 tiles -> 4096 blocks
    delta_kernel<<<4096, 256, 0, stream>>>(xdbl, W_dt, b_dt, delta);
    // K3: 262144 threads -> 1024 blocks
    scan_pass1<<<1024, 256, 0, stream>>>(delta, x, xdbl, A_log, aProd, hPart);
    // K4: 32768 threads -> 128 blocks
    scan_pass2<<<128, 256, 0, stream>>>(aProd, hPart, hIn);
    // K5: 262144 threads -> 1024 blocks
    scan_pass3<<<1024, 256, 0, stream>>>(delta, x, xdbl, A_log, D_param, hIn, y);
}